// GraphSAGERecommender_33131377721641
// MI455X (gfx1250) — compile-verified
//
#include <hip/hip_runtime.h>

typedef __attribute__((ext_vector_type(2))) float v2f;
typedef __attribute__((ext_vector_type(8))) float v8f;

#define EMB_D 64
#define TWO_D 128

// ---------------------------------------------------------------------------
// Zero the scatter-accumulation buffer (N*64 floats) with float4 stores.
// ---------------------------------------------------------------------------
__global__ void sage_zero_kernel(float* __restrict__ p, int n4) {
  int i = blockIdx.x * blockDim.x + threadIdx.x;
  if (i < n4) {
    ((float4*)p)[i] = make_float4(0.f, 0.f, 0.f, 0.f);
  }
}

// ---------------------------------------------------------------------------
// COO SpMM scatter: nbr[row] += val * x[col].  16 lanes per edge, one float4
// slice of the 64-wide embedding per lane.  Gather is a coalesced 256B burst
// per edge; scatter uses native fp32 global atomics (L2-resident table).
// ---------------------------------------------------------------------------
__global__ void sage_spmm_scatter(const int* __restrict__ rows,
                                  const int* __restrict__ cols,
                                  const float* __restrict__ vals,
                                  const float* __restrict__ x,
                                  float* __restrict__ out, int nEdges) {
  int idx = blockIdx.x * blockDim.x + threadIdx.x;
  int e = idx >> 4;            // edge id
  int c = (idx & 15) << 2;     // column quad: 0,4,...,60
  if (e >= nEdges) return;
  int r = rows[e];
  int s = cols[e];
  float v = vals[e];
  const float4 xv = *(const float4*)(x + (long long)s * EMB_D + c);
  float* o = out + (long long)r * EMB_D + c;
  unsafeAtomicAdd(o + 0, v * xv.x);
  unsafeAtomicAdd(o + 1, v * xv.y);
  unsafeAtomicAdd(o + 2, v * xv.z);
  unsafeAtomicAdd(o + 3, v * xv.w);
}

// ---------------------------------------------------------------------------
// Fused concat([h, nbr]) @ W + b, ReLU  via V_WMMA_F32_16X16X4_F32.
//
// Block = 256 threads = 8 wave32s; block tile = 32 rows x 64 cols.
//   wave -> (mTile = wave>>2 in {0,1}, nTile = wave&3 in {0..3})
//   each wave: one 16x16 f32 accumulator, K=128 in steps of 4 (32 WMMAs).
//
// Operand layout (ISA 7.12.2, f32 16x16x4):
//   A: lanes 0-15 hold M=lane with VGPR{0,1} = K={k,k+1};
//      lanes 16-31 hold M=lane-16 with K={k+2,k+3}.
//   B: lanes hold N, VGPR{0,1} = K pair selected by lane half (mirrors A).
//   C/D: VGPR v -> M = v + 8*half, N = lane&15.
//
// Safe for out == h (layer-2 in place): every block reads A only from the 32
// rows it writes, and __syncthreads() orders all A reads before any store.
// ---------------------------------------------------------------------------
__global__ void __launch_bounds__(256)
sage_gemm_relu(const float* __restrict__ h, const float* __restrict__ nbr,
               const float* __restrict__ W, const float* __restrict__ bias,
               float* __restrict__ out) {
  const int wave  = threadIdx.x >> 5;
  const int lane  = threadIdx.x & 31;
  const int half  = lane >> 4;        // 0: K={k,k+1}, 1: K={k+2,k+3}
  const int l     = lane & 15;
  const int mTile = wave >> 2;        // 0..1
  const int nTile = wave & 3;         // 0..3
  const int rowBase = blockIdx.x * 32 + mTile * 16;
  const int n       = nTile * 16 + l; // output column 0..63

  const float* aH = h   + (long long)(rowBase + l) * EMB_D; // K < 64 half
  const float* aN = nbr + (long long)(rowBase + l) * EMB_D; // K >= 64 half

  v8f acc = {};
#pragma unroll
  for (int k = 0; k < TWO_D; k += 4) {
    const int ka = k + half * 2;      // this lane-half's K base
    v2f a, b;
    if (k < EMB_D) {                  // compile-time after unroll; no straddle
      a.x = aH[ka];
      a.y = aH[ka + 1];
    } else {
      a.x = aN[ka - EMB_D];
      a.y = aN[ka - EMB_D + 1];
    }
    b.x = W[ka * EMB_D + n];
    b.y = W[(ka + 1) * EMB_D + n];
    acc = __builtin_amdgcn_wmma_f32_16x16x4_f32(
        /*neg_a=*/false, a, /*neg_b=*/false, b,
        /*c_mod=*/(short)0, acc, /*reuse_a=*/false, /*reuse_b=*/false);
  }

  // All A reads by this block are complete before any thread stores, which
  // makes in-place (out == h) execution race-free.
  __syncthreads();

  const float bv = bias[n];
  const int mOut = rowBase + half * 8;
#pragma unroll
  for (int v = 0; v < 8; ++v) {
    float r = acc[v] + bv;
    r = r > 0.f ? r : 0.f;
    out[(long long)(mOut + v) * EMB_D + n] = r;
  }
}

// ---------------------------------------------------------------------------
// Orchestration: two layers, nbr scratch in d_ws (N*64 floats = 51.2 MB).
// Layer-1 result lives in d_out; layer-2 overwrites it in place.
// ---------------------------------------------------------------------------
extern "C" void kernel_launch(void* const* d_in, const int* in_sizes, int n_in,
                              void* d_out, int out_size, void* d_ws, size_t ws_size,
                              hipStream_t stream) {
  const int*   rows = (const int*)d_in[0];
  const int*   cols = (const int*)d_in[1];
  const float* vals = (const float*)d_in[2];
  const float* emb  = (const float*)d_in[3];
  const float* W1   = (const float*)d_in[4];
  const float* b1   = (const float*)d_in[5];
  const float* W2   = (const float*)d_in[6];
  const float* b2   = (const float*)d_in[7];

  const int E = in_sizes[0];
  const int N = in_sizes[3] / EMB_D;   // 200000, divisible by 32

  float* outp = (float*)d_out;
  float* nbr  = (float*)d_ws;

  const int  n4 = (N * EMB_D) / 4;
  const dim3 zgrid((n4 + 255) / 256);
  const dim3 sgrid((E * 16 + 255) / 256);
  const dim3 ggrid(N / 32);
  const dim3 blk(256);

  // Layer 1: nbr = A @ emb ; out = relu([emb|nbr] @ W1 + b1)
  sage_zero_kernel<<<zgrid, blk, 0, stream>>>(nbr, n4);
  sage_spmm_scatter<<<sgrid, blk, 0, stream>>>(rows, cols, vals, emb, nbr, E);
  sage_gemm_relu<<<ggrid, blk, 0, stream>>>(emb, nbr, W1, b1, outp);

  // Layer 2: nbr = A @ out ; out = relu([out|nbr] @ W2 + b2)   (in place)
  sage_zero_kernel<<<zgrid, blk, 0, stream>>>(nbr, n4);
  sage_spmm_scatter<<<sgrid, blk, 0, stream>>>(rows, cols, vals, outp, nbr, E);
  sage_gemm_relu<<<ggrid, blk, 0, stream>>>(outp, nbr, W2, b2, outp);
}